// TopkLoss_15968688407351
// MI455X (gfx1250) — compile-verified
//
#include <hip/hip_runtime.h>
#include <math.h>

#define KTOP 5
#define BLOCK 256
#define TILE4 1024   // float4 slots per pipeline stage (16 KB)
#define LOG2E_F 1.4426950408889634f
#define LN2_F   0.6931471805599453f

static_assert(TILE4 % BLOCK == 0, "tile must be multiple of block");
#define JPT (TILE4 / BLOCK)   // async loads per lane per stage == 4

// s_wait_asynccnt <= n  (ASYNCcnt is per-wave, per-instruction, in-order)
#define ASYNC_WAIT_4() asm volatile("s_wait_asynccnt 4" ::: "memory")
#define ASYNC_WAIT_0() asm volatile("s_wait_asynccnt 0" ::: "memory")

// Issue one stage of async global->LDS b128 copies. Uniform EXEC: every lane
// issues JPT loads; tail lanes clamp the *global* index (LDS slots stay
// distinct, clamped slots are never consumed).
__device__ __forceinline__ void async_load_tile(unsigned long long gbase,
                                                int base4, int n4, int tid,
                                                float4 (&dst)[TILE4]) {
#pragma unroll
  for (int j = 0; j < JPT; ++j) {
    const int slot = j * BLOCK + tid;
    int gidx = base4 + slot;
    if (gidx > n4 - 1) gidx = n4 - 1;
    const unsigned goff = (unsigned)gidx * 16u;
    // low 32 bits of a flat LDS pointer == wave-relative LDS byte offset
    const unsigned lds = (unsigned)(uintptr_t)&dst[slot];
    asm volatile("global_load_async_to_lds_b128 %0, %1, %2"
                 :
                 : "v"(lds), "v"(goff), "s"(gbase)
                 : "memory");
  }
}

// One workgroup (8 wave32s) per row. Double-buffered async global->LDS
// pipeline; per-lane online softmax (base-2 domain) + top-k rank count.
__global__ __launch_bounds__(BLOCK) void topk_ce_row_kernel(
    const float* __restrict__ logits,
    const int*   __restrict__ target,
    float*       __restrict__ row_loss,
    int C) {
  __shared__ float4 buf[2][TILE4];

  const int row = blockIdx.x;
  const float* __restrict__ rp = logits + (size_t)row * (size_t)C;
  const int tgt = target[row];
  const float v = rp[tgt];

  const int tid = threadIdx.x;
  const int n4 = C >> 2;                        // 8000 for C=32000
  const int nstage = (n4 + TILE4 - 1) / TILE4;  // 8 stages
  const unsigned long long gbase = (unsigned long long)(uintptr_t)rp;

  float m = -INFINITY;  // running max of x*log2(e)
  float s = 0.0f;       // running sum of 2^(t - m)
  int   cnt = 0;        // rank: #(x>v) + #(x==v && idx<tgt)

  // prologue: stage 0 in flight
  async_load_tile(gbase, 0, n4, tid, buf[0]);

  for (int st = 0; st < nstage; ++st) {
    const int b = st & 1;
    const int base4 = st * TILE4;

    if (st + 1 < nstage) {
      async_load_tile(gbase, base4 + TILE4, n4, tid, buf[b ^ 1]);
      ASYNC_WAIT_4();   // only next stage's 4 loads may remain outstanding
    } else {
      ASYNC_WAIT_0();   // drain: last stage's data ready
    }

    const int rem = n4 - base4;
    const int v4 = rem < TILE4 ? rem : TILE4;   // valid slots this stage
#pragma unroll
    for (int j = 0; j < JPT; ++j) {
      const int slot = j * BLOCK + tid;
      if (slot < v4) {
        float4 x4 = buf[b][slot];               // ds_load_b128
        const int gi0 = (base4 + slot) << 2;
        float xs[4] = {x4.x, x4.y, x4.z, x4.w};
#pragma unroll
        for (int c = 0; c < 4; ++c) {
          float x = xs[c];
          int idx = gi0 + c;
          cnt += ((x > v) || (x == v && idx < tgt)) ? 1 : 0;
          float t = x * LOG2E_F;
          if (t > m) {            // rare: new running max
            s = s * exp2f(m - t) + 1.0f;
            m = t;
          } else {                // common: single v_exp_f32
            s += exp2f(t - m);
          }
        }
      }
    }
  }

  // wave32 butterfly reduction of (m, s, cnt)
#pragma unroll
  for (int off = 16; off > 0; off >>= 1) {
    float om = __shfl_xor(m, off, 32);
    float os = __shfl_xor(s, off, 32);
    int   oc = __shfl_xor(cnt, off, 32);
    float M = fmaxf(m, om);
    s = s * exp2f(m - M) + os * exp2f(om - M);
    m = M;
    cnt += oc;
  }

  __shared__ float sm_m[BLOCK / 32];
  __shared__ float sm_s[BLOCK / 32];
  __shared__ int   sm_c[BLOCK / 32];
  const int wave = tid >> 5;
  const int lane = tid & 31;
  if (lane == 0) { sm_m[wave] = m; sm_s[wave] = s; sm_c[wave] = cnt; }
  __syncthreads();

  if (tid == 0) {
    float M = sm_m[0], S = sm_s[0];
    int   Cn = sm_c[0];
#pragma unroll
    for (int w = 1; w < BLOCK / 32; ++w) {
      float om = sm_m[w], os = sm_s[w];
      float NM = fmaxf(M, om);
      S = S * exp2f(M - NM) + os * exp2f(om - NM);
      M = NM;
      Cn += sm_c[w];
    }
    float lse = (M + log2f(S)) * LN2_F;  // back to natural log
    float ce  = lse - v;
    row_loss[row] = (Cn < KTOP) ? 0.0f : ce;   // in top-K -> masked
  }
}

// Deterministic final reduction: fixed-order strided accumulate + LDS tree.
__global__ __launch_bounds__(BLOCK) void topk_ce_reduce_kernel(
    const float* __restrict__ row_loss, float* __restrict__ out, int n) {
  __shared__ float sm[BLOCK];
  float acc = 0.0f;
  for (int i = threadIdx.x; i < n; i += BLOCK) acc += row_loss[i];
  sm[threadIdx.x] = acc;
  __syncthreads();
#pragma unroll
  for (int sft = BLOCK / 2; sft > 0; sft >>= 1) {
    if (threadIdx.x < sft) sm[threadIdx.x] += sm[threadIdx.x + sft];
    __syncthreads();
  }
  if (threadIdx.x == 0) out[0] = sm[0] / (float)n;
}

extern "C" void kernel_launch(void* const* d_in, const int* in_sizes, int n_in,
                              void* d_out, int out_size, void* d_ws, size_t ws_size,
                              hipStream_t stream) {
  const float* logits = (const float*)d_in[0];   // [B*T, C] fp32
  const int*   target = (const int*)d_in[1];     // [B*T] int
  float* out = (float*)d_out;                    // scalar fp32
  float* ws  = (float*)d_ws;                     // per-row masked CE

  const int N = in_sizes[1];                     // 8192 rows
  const int C = in_sizes[0] / N;                 // 32000 classes

  topk_ce_row_kernel<<<N, BLOCK, 0, stream>>>(logits, target, ws, C);
  topk_ce_reduce_kernel<<<1, BLOCK, 0, stream>>>(ws, out, N);
}